// LML_33698313404564
// MI455X (gfx1250) — compile-verified
//
#include <hip/hip_runtime.h>
#include <hip/hip_bf16.h>
#include <cstdint>

#define COLS 1024
#define CHUNKS (COLS / 16)       // 64 WMMA tiles per row per sweep
#define ROWS_PER_BLOCK 4
#define EPSV 1e-4f
#define MAX_ITER 20

typedef __attribute__((ext_vector_type(2))) float v2f;
typedef __attribute__((ext_vector_type(8))) float v8f;
typedef __attribute__((ext_vector_type(4))) unsigned int v4u;
typedef __attribute__((ext_vector_type(4))) int v4i;
typedef __attribute__((ext_vector_type(8))) int v8i;

__device__ __forceinline__ float fast_tanh(float v) {
#if __has_builtin(__builtin_amdgcn_tanhf)
  return __builtin_amdgcn_tanhf(v);              // v_tanh_f32: 1 trans op
#else
  float e = __builtin_amdgcn_exp2f(-2.885390081777927f * v);  // e^(-2v)
  return (1.0f - e) * __builtin_amdgcn_rcpf(1.0f + e);
#endif
}

__global__ __launch_bounds__(ROWS_PER_BLOCK * 32)
void lml_kernel(const float* __restrict__ x, float* __restrict__ y, int rows) {
  __shared__ float tile[ROWS_PER_BLOCK * COLS];
  const int lane = threadIdx.x & 31;
  const int wave = threadIdx.x >> 5;
  const int rowBase = blockIdx.x * ROWS_PER_BLOCK;
  const int rowsHere = min(ROWS_PER_BLOCK, rows - rowBase);

  // ---- TDM: async DMA of the rowsHere x 1024 f32 tile into LDS ----
  if (wave == 0) {
    uint64_t gaddr = (uint64_t)(uintptr_t)(x + (size_t)rowBase * COLS);
    uint32_t laddr = (uint32_t)(uintptr_t)&tile[0];   // flat low 32 = LDS offset
    v4u g0; v8i g1;
    v4i g2 = {0, 0, 0, 0};
    v4i g3 = {0, 0, 0, 0};
    v8i g4 = {0, 0, 0, 0, 0, 0, 0, 0};
    g0[0] = 1u;                                  // count=1, user descriptor
    g0[1] = laddr;                               // lds_addr (bytes)
    g0[2] = (uint32_t)gaddr;                     // global_addr[31:0]
    g0[3] = ((uint32_t)(gaddr >> 32) & 0x01FFFFFFu) | 0x80000000u; // addr[56:32], type=2
    g1[0] = (int)(2u << 16);                     // data_size = 4 bytes
    g1[1] = (int)(((uint32_t)COLS & 0xFFFFu) << 16);              // tensor_dim0 lo -> [63:48]
    g1[2] = (int)(((uint32_t)COLS >> 16) |
                  (((uint32_t)rowsHere & 0xFFFFu) << 16));        // td0 hi | tensor_dim1 lo
    g1[3] = (int)(((uint32_t)rowsHere >> 16) |
                  ((uint32_t)COLS << 16));                        // td1 hi | tile_dim0
    g1[4] = rowsHere;                            // tile_dim1 (tile_dim2 = 0)
    g1[5] = COLS;                                // tensor_dim0_stride lo32 (elements)
    g1[6] = 0; g1[7] = 0;
    __builtin_amdgcn_tensor_load_to_lds(g0, g1, g2, g3, g4, 0);
    __builtin_amdgcn_s_wait_tensorcnt(0);
  }
  __syncthreads();

  if (wave >= rowsHere) return;
  const int row = rowBase + wave;
  const float* lrow = tile + wave * COLS;
  const float4* lrow4 = (const float4*)lrow;

  // ---- bracket init from row min/max: f(-max-7) < 0 < f(-min+7) ----
  float vmin = 3.0e38f, vmax = -3.0e38f;
#pragma unroll
  for (int k = 0; k < COLS / 128; ++k) {
    float4 v = lrow4[lane + 32 * k];
    vmin = fminf(vmin, fminf(fminf(v.x, v.y), fminf(v.z, v.w)));
    vmax = fmaxf(vmax, fmaxf(fmaxf(v.x, v.y), fmaxf(v.z, v.w)));
  }
#pragma unroll
  for (int m = 16; m; m >>= 1) {
    vmin = fminf(vmin, __shfl_xor(vmin, m, 32));
    vmax = fmaxf(vmax, __shfl_xor(vmax, m, 32));
  }
  float lo = -vmax - 7.0f;
  float hi = -vmin + 7.0f;

  // ---- preload x/2 into WMMA 16x16x4 A-matrix layout (lanes 0-15 = rows) ----
  float xh[CHUNKS];
#pragma unroll
  for (int t = 0; t < CHUNKS; ++t)
    xh[t] = (lane < 16) ? 0.5f * lrow[t * 16 + (lane & 15)] : 0.0f;
  const float aones = (lane < 16) ? 1.0f : 0.0f;   // A column 1; cols 2,3 = 0

  // ---- 16-way multisection with early exit (== reference `active` gating) ----
  for (int it = 0; it < MAX_ITER && (hi - lo) > EPSV; ++it) {
    const float r = hi - lo;
    const float step = r * (1.0f / 15.0f);
    const float nuh = 0.5f * (lo + step * (float)(lane & 15));
    v2f b;
    b.x = (lane < 16) ? 1.0f : nuh;   // B row0 = 1s (either striping)
    b.y = (lane < 16) ? nuh : 0.0f;   // B row1 = nu/2 (either striping)
    v8f acc = {0.f, 0.f, 0.f, 0.f, 0.f, 0.f, 0.f, 0.f};
#pragma unroll
    for (int t = 0; t < CHUNKS; ++t) {
      v2f a; a.x = xh[t]; a.y = aones;
      v8f c = {0.f, 0.f, 0.f, 0.f, 0.f, 0.f, 0.f, 0.f};
      // D[m][n] = (x_m + nu_n)/2 : outer-sum on the matrix pipe
      v8f d = __builtin_amdgcn_wmma_f32_16x16x4_f32(false, a, false, b,
                                                    (short)0, c, false, false);
      // Phase 1: 8 independent TRANS ops back-to-back (no hazard v_nops:
      // each tanh fills the previous tanh's 1-op hazard slot)
      float tt[8];
#pragma unroll
      for (int i = 0; i < 8; ++i) tt[i] = fast_tanh(d[i]);
      // Phase 2: 8 accumulate adds (dual-issuable VALU)
#pragma unroll
      for (int i = 0; i < 8; ++i) acc[i] += tt[i];
    }
    // per-candidate total T(n): 8 D-rows per lane half, + partner half
    float s = ((acc[0] + acc[1]) + (acc[2] + acc[3])) +
              ((acc[4] + acc[5]) + (acc[6] + acc[7]));
    s += __shfl_xor(s, 16, 32);                 // lane holds T for n = lane&15
    // f(n) < 0  <=>  512 + T/2 - 5 < 0  <=>  T < -1014
    float isneg = ((lane < 16) && (s < -1014.0f)) ? 1.0f : 0.0f;
#pragma unroll
    for (int m = 16; m; m >>= 1) isneg += __shfl_xor(isneg, m, 32);
    int ilo = max((int)isneg - 1, 0);
    int ihi = min(ilo + 1, 15);
    float nlo = lo + step * (float)ilo;
    float nhi = lo + step * (float)ihi;
    lo = nlo; hi = nhi;
  }
  const float nu = lo + 0.5f * (hi - lo);

  // ---- streaming output: y = sigmoid(x + nu), LDS -> global, float4 ----
  float4* gy = (float4*)(y + (size_t)row * COLS);
#pragma unroll
  for (int k = 0; k < COLS / 128; ++k) {
    float4 v = lrow4[lane + 32 * k];
    float4 o;
    o.x = 1.0f / (1.0f + __expf(-(v.x + nu)));
    o.y = 1.0f / (1.0f + __expf(-(v.y + nu)));
    o.z = 1.0f / (1.0f + __expf(-(v.z + nu)));
    o.w = 1.0f / (1.0f + __expf(-(v.w + nu)));
    gy[lane + 32 * k] = o;
  }
}

extern "C" void kernel_launch(void* const* d_in, const int* in_sizes, int n_in,
                              void* d_out, int out_size, void* d_ws, size_t ws_size,
                              hipStream_t stream) {
  (void)n_in; (void)out_size; (void)d_ws; (void)ws_size;
  const float* x = (const float*)d_in[0];
  float* y = (float*)d_out;
  const int rows = in_sizes[0] / COLS;
  const int blocks = (rows + ROWS_PER_BLOCK - 1) / ROWS_PER_BLOCK;
  lml_kernel<<<blocks, ROWS_PER_BLOCK * 32, 0, stream>>>(x, y, rows);
}